// CgpHmmCell_onedim_29334626632212
// MI455X (gfx1250) — compile-verified
//
#include <hip/hip_runtime.h>
#include <hip/hip_bf16.h>
#include <math.h>

typedef float v2f __attribute__((ext_vector_type(2)));
typedef float v8f __attribute__((ext_vector_type(8)));

#define N_STATES 25
#define SPAD     32           // padded state count (multiple of 16 for WMMA)
#define ALPHA3   216          // 6^3 emission contexts
#define BATCH    512
#define TLEN     2048

// workspace layout (bytes)
#define WS_A_OFF    0                        // 32*32 f32, A[in*32 + out]
#define WS_B_OFF    4096                     // 25*216 f32
#define WS_I_OFF    (4096 + 21632)           // 25 f32 (21600 rounded up)
#define WS_FLAT_OFF (4096 + 21632 + 128)     // 512*2048 u8

// ---------------------------------------------------------------------------
// Kernel 1: build A (25x25 softmax, padded to 32x32), B (25x216 softmax from
// the emission index spec), I (softmax of first 9 init logits).
// ---------------------------------------------------------------------------
__global__ void setup_tables(const float* __restrict__ wA,
                             const float* __restrict__ wB,
                             const float* __restrict__ wI,
                             float* __restrict__ ws) {
  __shared__ float L[N_STATES * ALPHA3];   // reused: A logits (1024) then B logits (5400)
  float* Apad = ws + (WS_A_OFF / 4);
  float* Bout = ws + (WS_B_OFF / 4);
  float* Iout = ws + (WS_I_OFF / 4);
  const float NEG = -1e30f;
  const int tid = threadIdx.x;

  // ---------------- A logits (thread 0, tiny) ----------------
  if (tid == 0) {
    for (int i = 0; i < SPAD * SPAD; ++i) L[i] = NEG;
    const float* w = wA;
    #define ADD(i, j, v) L[(i) * SPAD + (j)] = (v)
    ADD(0, 0, 1.f - w[0]); ADD(0, 1, w[0]);
    ADD(1, 2, 1.f); ADD(2, 3, 1.f);
    for (int i = 0; i < 2; ++i) ADD(3 + 3 * i, 4 + 3 * i, w[1 + i]);
    for (int i = 0; i < 2; ++i) ADD(4 + 3 * i, 5 + 3 * i, 1.f);
    for (int i = 0; i < 2; ++i) ADD(5 + 3 * i, 6 + 3 * i, 1.f);
    for (int i = 0; i < 3; ++i) ADD(3 + 3 * i, 14 + 3 * i, w[3 + i]);
    ADD(9, 10, 1.f - w[5]);
    for (int i = 0; i < 3; ++i) ADD(14 + 3 * i, 15 + 3 * i, 1.f);
    for (int i = 0; i < 3; ++i) ADD(15 + 3 * i, 16 + 3 * i, 1.f);
    for (int i = 0; i < 3; ++i) ADD(16 + 3 * i, 4 + 3 * i, w[6 + i]);
    for (int i = 0; i < 3; ++i) ADD(16 + 3 * i, 14 + 3 * i, 1.f - w[6 + i]);
    const float w9 = w[9];
    ADD(3, 7,  1.f - w9 * w9);        // (7-3)//3 = 1
    ADD(3, 10, 1.f - w9 * w9 * w9);   // (10-3)//3 = 2
    ADD(6, 10, 1.f - w9 * w9);        // (10-6)//3 = 1
    ADD(10, 11, 1.f); ADD(11, 12, 1.f); ADD(12, 13, 1.f);
    ADD(13, 13, 1.f); ADD(13, 23, 1.f);
    ADD(23, 23, 1.f); ADD(23, 24, 1.f); ADD(24, 24, 1.f);
    #undef ADD
  }
  __syncthreads();
  if (tid < SPAD) {
    const int r = tid;
    if (r < N_STATES) {
      float m = NEG;
      for (int j = 0; j < SPAD; ++j) m = fmaxf(m, L[r * SPAD + j]);
      float s = 0.f;
      for (int j = 0; j < SPAD; ++j) {
        float x = L[r * SPAD + j];
        if (x > NEG * 0.5f) s += __expf(x - m);
      }
      for (int j = 0; j < SPAD; ++j) {
        float x = L[r * SPAD + j];
        Apad[r * SPAD + j] = (x > NEG * 0.5f) ? __expf(x - m) / s : 0.f;
      }
    } else {
      for (int j = 0; j < SPAD; ++j) Apad[r * SPAD + j] = 0.f;
    }
  }
  __syncthreads();

  // ---------------- B logits via the emission index spec ----------------
  if (tid == 0) {
    for (int i = 0; i < N_STATES * ALPHA3; ++i) L[i] = NEG;
    // {state, mask_pos0, mask_pos1, mask_pos2, trainable}
    // bit i of mask = symbol i (0..5) allowed; ascending symbol order
    // matches itertools.product order (CODE values < appended 4 < X=5 cases).
    static const unsigned char S[29][5] = {
      { 0, 0x1F, 0x1F, 0x0F, 1},  // 'N'
      { 1, 0x1F, 0x1F, 0x01, 1},  // 'A'
      { 2, 0x1F, 0x11, 0x08, 1},  // 'AT'
      { 3, 0x11, 0x18, 0x04, 0},  // 'ATG' (fixed)
      { 4, 0x18, 0x14, 0x0F, 1},  // 'ATGN' -> TGN
      { 5, 0x14, 0x1F, 0x0F, 1},  // 'ATGNN' -> GNN
      { 6, 0x1F, 0x1F, 0x0F, 1},  // codon 'N'
      { 7, 0x1F, 0x1F, 0x0F, 1},
      { 8, 0x1F, 0x1F, 0x0F, 1},
      { 9, 0x1F, 0x1F, 0x0F, 1},
      {10, 0x0F, 0x0F, 0x08, 1},  // 'T', xp=2 (no 4s)
      {11, 0x0F, 0x08, 0x01, 1},  // 'TA'
      {11, 0x0F, 0x08, 0x04, 1},  // 'TG'
      {12, 0x08, 0x01, 0x01, 0},  // 'TAA' (fixed)
      {12, 0x08, 0x01, 0x04, 0},  // 'TAG' (fixed)
      {12, 0x08, 0x04, 0x01, 0},  // 'TGA' (fixed)
      {13, 0x0F, 0x0F, 0x0F, 1},  // 'N', xp=2
      {14, 0x1F, 0x1F, 0x0F, 1},  // intron 'N' states
      {15, 0x1F, 0x1F, 0x0F, 1},
      {16, 0x1F, 0x1F, 0x0F, 1},
      {17, 0x1F, 0x1F, 0x0F, 1},
      {18, 0x1F, 0x1F, 0x0F, 1},
      {19, 0x1F, 0x1F, 0x0F, 1},
      {20, 0x1F, 0x1F, 0x0F, 1},
      {21, 0x1F, 0x1F, 0x0F, 1},
      {22, 0x1F, 0x1F, 0x0F, 1},
      {23, 0x0F, 0x0F, 0x20, 1},  // 'X'
      {23, 0x0F, 0x20, 0x20, 1},  // 'XX'
      {24, 0x20, 0x20, 0x20, 0},  // 'XXX' (fixed)
    };
    int k = 0;
    for (int sp = 0; sp < 29; ++sp) {
      const int st = S[sp][0];
      const int m0 = S[sp][1], m1 = S[sp][2], m2 = S[sp][3];
      const int tr = S[sp][4];
      for (int c0 = 0; c0 < 6; ++c0) {
        if (!((m0 >> c0) & 1)) continue;
        for (int c1 = 0; c1 < 6; ++c1) {
          if (!((m1 >> c1) & 1)) continue;
          for (int c2 = 0; c2 < 6; ++c2) {
            if (!((m2 >> c2) & 1)) continue;
            if (c0 != 4 && c1 == 4) continue;        // x-precede filter
            float v = tr ? wB[k] : 1.f;
            if (tr) ++k;
            L[st * ALPHA3 + c0 * 36 + c1 * 6 + c2] = v; // last write wins
          }
        }
      }
    }
  }
  __syncthreads();
  if (tid < N_STATES) {
    const int r = tid;
    float m = NEG;
    for (int j = 0; j < ALPHA3; ++j) m = fmaxf(m, L[r * ALPHA3 + j]);
    float s = 0.f;
    for (int j = 0; j < ALPHA3; ++j) {
      float x = L[r * ALPHA3 + j];
      if (x > NEG * 0.5f) s += __expf(x - m);
    }
    for (int j = 0; j < ALPHA3; ++j) {
      float x = L[r * ALPHA3 + j];
      Bout[r * ALPHA3 + j] = (x > NEG * 0.5f) ? __expf(x - m) / s : 0.f;
    }
  }
  __syncthreads();

  // ---------------- I ----------------
  if (tid == 0) {
    const int n_init = 9;  // 3 + 3*N_CODONS
    float m = -1e30f;
    for (int i = 0; i < n_init; ++i) m = fmaxf(m, wI[i]);
    float s = 0.f;
    for (int i = 0; i < n_init; ++i) s += __expf(wI[i] - m);
    for (int i = 0; i < N_STATES; ++i)
      Iout[i] = (i < n_init) ? __expf(wI[i] - m) / s : 0.f;
  }
}

// ---------------------------------------------------------------------------
// Kernel 2: precompute flat emission context index per (seq, t) as u8.
// flat = tp[t-2]*36 + tp[t-1]*6 + tp[t], padded with 4 ('N') — max 171 < 256.
// ---------------------------------------------------------------------------
__global__ void flat_kernel(const int* __restrict__ tokens,
                            unsigned char* __restrict__ flat) {
  int idx = blockIdx.x * blockDim.x + threadIdx.x;
  if (idx >= BATCH * TLEN) return;
  int b = idx / TLEN, t = idx - b * TLEN;
  int t2 = tokens[b * TLEN + t];
  int t1 = (t >= 1) ? tokens[b * TLEN + t - 1] : 4;
  int t0 = (t >= 2) ? tokens[b * TLEN + t - 2] : 4;
  flat[idx] = (unsigned char)(t0 * 36 + t1 * 6 + t2);
}

// ---------------------------------------------------------------------------
// Kernel 3: forward scan. One wave per 16 sequences (32 waves total).
// Transposed recurrence: alpha^T <- A^T @ alpha^T via V_WMMA_F32_16X16X4_F32,
// so D layout = (state rows in VGPR index, sequences across lanes):
//   normalization = register sum + one shfl_xor(16); E multiply = LDS gather.
// A^T is pinned in 32 VGPRs in WMMA A-layout for the whole scan.
// ---------------------------------------------------------------------------
__global__ __launch_bounds__(128, 1)
void hmm_forward(const float* __restrict__ ws_f,
                 const unsigned char* __restrict__ flat,
                 float* __restrict__ out) {
  __shared__ float Bsh[SPAD * ALPHA3];   // 32x216 f32 = 27648 B (rows >=25 zero)
  const float* Apad = ws_f + (WS_A_OFF / 4);
  const float* Bws  = ws_f + (WS_B_OFF / 4);
  const float* Iws  = ws_f + (WS_I_OFF / 4);

  for (int i = threadIdx.x; i < SPAD * ALPHA3; i += blockDim.x) {
    int r = i / ALPHA3;
    Bsh[i] = (r < N_STATES) ? Bws[i] : 0.f;
  }
  __syncthreads();

  const int lane    = threadIdx.x & 31;
  const int wave    = threadIdx.x >> 5;
  const int gwave   = blockIdx.x * 4 + wave;   // 0..31
  const int seqBase = gwave * 16;
  const int q       = lane & 15;               // sequence within tile
  const int hi      = lane >> 4;               // half-wave flag

  // Pin A^T in WMMA A-layout: tile m covers out-states 16m..16m+15,
  // chunk k covers in-states 4k..4k+3. value = A[in][out].
  v2f a0[8], a1[8];
  {
    const int kofs = hi ? 2 : 0;
    #pragma unroll
    for (int k = 0; k < 8; ++k) {
      const int kb = 4 * k + kofs;
      a0[k].x = Apad[kb * SPAD + q];
      a0[k].y = Apad[(kb + 1) * SPAD + q];
      a1[k].x = Apad[kb * SPAD + 16 + q];
      a1[k].y = Apad[(kb + 1) * SPAD + 16 + q];
    }
  }

  const unsigned char* fp = flat + (size_t)(seqBase + q) * TLEN;

  // ---- t = 0: alpha = I * E[0], normalize ----
  float alpha0[8], alpha1[8];  // D layout: alpha0 = states 0..15, alpha1 = 16..31
  float ll;
  {
    int f0 = (lane < 16) ? (int)fp[0] : 0;
    f0 = __shfl(f0, q, 32);
    float psum = 0.f;
    #pragma unroll
    for (int r = 0; r < 8; ++r) {
      const int s0 = r + hi * 8;
      const int s1 = s0 + 16;
      const float i0 = (s0 < N_STATES) ? Iws[s0] : 0.f;
      const float i1 = (s1 < N_STATES) ? Iws[s1] : 0.f;
      alpha0[r] = i0 * Bsh[s0 * ALPHA3 + f0];
      alpha1[r] = i1 * Bsh[s1 * ALPHA3 + f0];
      psum += alpha0[r] + alpha1[r];
    }
    const float ssum = psum + __shfl_xor(psum, 16, 32);
    const float inv = 1.f / ssum;
    #pragma unroll
    for (int r = 0; r < 8; ++r) { alpha0[r] *= inv; alpha1[r] *= inv; }
    ll = __logf(ssum);
  }

  // ---- scan t = 1..T-1 ----
  for (int t = 1; t < TLEN; ++t) {
    v8f acc0 = {}; v8f acc1 = {};
    #pragma unroll
    for (int k = 0; k < 8; ++k) {
      // Re-marshal alpha (D layout) into WMMA B-layout for in-states 4k..4k+3.
      const int base = (k & 3) * 4;       // 0,4,8,12 within source tile
      const int r0 = base & 7;            // source VGPR index
      const int h  = base >> 3;           // which lane-half holds these states
      const float e0 = (k < 4) ? alpha0[r0]     : alpha1[r0];
      const float e1 = (k < 4) ? alpha0[r0 + 1] : alpha1[r0 + 1];
      const float e2 = (k < 4) ? alpha0[r0 + 2] : alpha1[r0 + 2];
      const float e3 = (k < 4) ? alpha0[r0 + 3] : alpha1[r0 + 3];
      v2f b;
      if (h == 0) {
        const float x2 = __shfl_xor(e2, 16, 32);
        const float x3 = __shfl_xor(e3, 16, 32);
        b.x = (lane < 16) ? e0 : x2;      // K=0 rows low-half, K=2 rows high-half
        b.y = (lane < 16) ? e1 : x3;      // K=1 / K=3
      } else {
        const float x0 = __shfl_xor(e0, 16, 32);
        const float x1 = __shfl_xor(e1, 16, 32);
        b.x = (lane < 16) ? x0 : e2;
        b.y = (lane < 16) ? x1 : e3;
      }
      acc0 = __builtin_amdgcn_wmma_f32_16x16x4_f32(false, a0[k], false, b,
                                                   (short)0, acc0, false, false);
      acc1 = __builtin_amdgcn_wmma_f32_16x16x4_f32(false, a1[k], false, b,
                                                   (short)0, acc1, false, false);
    }

    // E multiply + normalization (sequence = lane&15)
    int f = (lane < 16) ? (int)fp[t] : 0;
    f = __shfl(f, q, 32);
    float psum = 0.f;
    #pragma unroll
    for (int r = 0; r < 8; ++r) {
      const int s0 = r + hi * 8;
      const int s1 = s0 + 16;
      const float v0 = acc0[r] * Bsh[s0 * ALPHA3 + f];
      const float v1 = acc1[r] * Bsh[s1 * ALPHA3 + f];
      alpha0[r] = v0; alpha1[r] = v1;
      psum += v0 + v1;
    }
    const float ssum = psum + __shfl_xor(psum, 16, 32);
    const float inv = 1.f / ssum;
    #pragma unroll
    for (int r = 0; r < 8; ++r) { alpha0[r] *= inv; alpha1[r] *= inv; }
    ll += __logf(ssum);
  }

  if (lane < 16) out[seqBase + q] = ll;
}

// ---------------------------------------------------------------------------
extern "C" void kernel_launch(void* const* d_in, const int* in_sizes, int n_in,
                              void* d_out, int out_size, void* d_ws, size_t ws_size,
                              hipStream_t stream) {
  const float* wA     = (const float*)d_in[0];   // transition_kernel (10)
  const float* wB     = (const float*)d_in[1];   // emission_kernel (5400)
  const float* wI     = (const float*)d_in[2];   // init_kernel (25)
  const int*   tokens = (const int*)d_in[3];     // (512, 2048)
  float* ws = (float*)d_ws;
  unsigned char* flatbuf = (unsigned char*)d_ws + WS_FLAT_OFF;

  setup_tables<<<1, 64, 0, stream>>>(wA, wB, wI, ws);
  flat_kernel<<<(BATCH * TLEN + 255) / 256, 256, 0, stream>>>(tokens, flatbuf);
  hmm_forward<<<8, 128, 0, stream>>>(ws, flatbuf, (float*)d_out);
}